// WTConv1d_73108933313161
// MI455X (gfx1250) — compile-verified
//
// WTConv1d (3-level Haar + depthwise convs) fully fused single-pass kernel
// for AMD MI455X / gfx1250 (CDNA5, wave32).
//
// Rationale (compile-only; reasoned from MI455X specs):
//  * Workload is depthwise -> no channel reduction -> WMMA not applicable.
//  * Memory-bound: 256 MB in + 256 MB out minimum => ~22 us at 23.3 TB/s.
//    Therefore: ONE fused pass, all intermediates live in LDS (320 KB/WGP).
//  * One row (16384 f32) per workgroup; per-row state = 176 KB; plus a
//    second 64 KB X buffer for double-buffered DMA prefetch = 240 KB LDS.
//  * CDNA5 path: TDM tensor_load_to_lds (one descriptor per wave, 4 KB each,
//    TENSORcnt + s_wait_tensorcnt) overlaps the next row's HBM load with the
//    current row's LDS compute; fallback = global_load_async_to_lds_b128.

#include <hip/hip_runtime.h>

#define LDS_AS __attribute__((address_space(3)))
#define GLB_AS __attribute__((address_space(1)))

#ifndef __has_builtin
#define __has_builtin(x) 0
#endif

#define WT_STR2(x) #x
#define WT_STR(x) WT_STR2(x)

#if __has_builtin(__builtin_amdgcn_tensor_load_to_lds) && \
    __has_builtin(__builtin_amdgcn_s_wait_tensorcnt)
#define WT_TDM 1
#else
#define WT_TDM 0
#endif
#if __has_builtin(__builtin_amdgcn_global_load_async_to_lds_b128)
#define WT_ASYNC 1
#else
#define WT_ASYNC 0
#endif

#if WT_TDM
#pragma message("WTCONV: TDM tensor_load_to_lds path, __clang_major__ = " WT_STR(__clang_major__))
#elif WT_ASYNC
#pragma message("WTCONV: global_load_async_to_lds_b128 path")
#else
#pragma message("WTCONV: sync copy fallback path")
#endif

typedef int          wt_v4i   __attribute__((vector_size(16)));
typedef unsigned int wt_u32x4 __attribute__((vector_size(16)));
typedef int          wt_i32x4 __attribute__((vector_size(16)));
typedef int          wt_i32x8 __attribute__((vector_size(32)));
typedef float        wt_f32x4 __attribute__((vector_size(16)));  // native vec for NT store

// ---- problem constants (match reference setup_inputs) ----
constexpr int kL  = 16384;   // row length
constexpr int kC  = 256;     // channels
constexpr int kT  = 512;     // threads per block (16 wave32)
constexpr int kL1 = kL / 2;  // 8192
constexpr int kL2 = kL / 4;  // 4096
constexpr int kL3 = kL / 8;  // 2048
constexpr int kWaves     = kT / 32;      // 16
constexpr int kWaveChunk = kL / kWaves;  // 1024 floats = 4 KB per TDM descriptor

// ---- LDS layout (float offsets) ----
constexpr int oX0  = 0;
constexpr int oX1  = kL;             // double-buffered raw row
constexpr int oA   = 2 * kL;         // level-1 low  (ylow1 -> ll1 -> ll1+nll1)
constexpr int oB   = oA + kL1;       // level-1 high (yhigh1 -> hh1)
constexpr int oC2L = oB + kL1;       // level-2 low
constexpr int oC2H = oC2L + kL2;     // level-2 high
constexpr int oD3L = oC2H + kL2;     // level-3 low
constexpr int oD3H = oD3L + kL3;     // level-3 high
constexpr int kLdsFloats = oD3H + kL3;   // 61440 floats = 245760 B (240 KB)

// ---------------------------------------------------------------------------
__device__ __forceinline__ unsigned wt_lds_offset(const void* p) {
  // addrspacecast generic->LDS yields the 32-bit workgroup-relative offset.
  return (unsigned)(unsigned long long)(LDS_AS const void*)p;
}

// Load one 16384-float row global -> LDS.
//  TDM path: each of the 16 waves issues ONE tensor-DMA descriptor covering
//  its contiguous 4 KB slice (TDM ignores EXEC; per-wave slicing means each
//  wave legitimately issues exactly one op and waits its own TENSORcnt).
__device__ __forceinline__ void row_load(float* lds_dst, const float* gsrc, int tid) {
#if WT_TDM
  const int wave = tid >> 5;  // wave-uniform
  const unsigned lds_off = wt_lds_offset(lds_dst) + (unsigned)wave * (kWaveChunk * 4u);
  const unsigned long long ga =
      (unsigned long long)(size_t)(gsrc + wave * kWaveChunk);

  // D# group 0 (ISA 8.3): count=1 | lds_addr | global_addr[56:0] | type=2
  wt_u32x4 g0;
  g0[0] = 1u;
  g0[1] = lds_off;
  g0[2] = (unsigned)ga;
  g0[3] = ((unsigned)(ga >> 32) & 0x01FFFFFFu) | 0x80000000u;

  // D# group 1 (ISA 8.4): data_size=4B; tensor_dim0=tile_dim0=kWaveChunk;
  // tensor_dim1=1; tile_dim1/2=0 (1-D tile); stride0=kWaveChunk.
  wt_i32x8 g1;
  g1[0] = 2 << 16;                               // data_size = 4 bytes
  g1[1] = (int)((kWaveChunk & 0xffff) << 16);    // tensor_dim0[15:0]
  g1[2] = (int)((kWaveChunk >> 16) | (1 << 16)); // tensor_dim0[31:16] | dim1 lo
  g1[3] = (int)((kWaveChunk & 0xffff) << 16);    // dim1 hi | tile_dim0
  g1[4] = 0;                                     // tile_dim1=0, tile_dim2=0
  g1[5] = kWaveChunk;                            // tensor_dim0_stride[31:0]
  g1[6] = 0;
  g1[7] = 0;

  wt_i32x4 z4 = {0, 0, 0, 0};                    // groups 2/3: <=2-D tensor
#if __clang_major__ >= 23
  wt_i32x8 z8 = {0, 0, 0, 0, 0, 0, 0, 0};
  __builtin_amdgcn_tensor_load_to_lds(g0, g1, z4, z4, z8, 0);
#else
  __builtin_amdgcn_tensor_load_to_lds(g0, g1, z4, z4, 0);
#endif

#elif WT_ASYNC
  float4* d4 = (float4*)lds_dst;
  const float4* s4 = (const float4*)gsrc;
#pragma unroll
  for (int j = 0; j < kL / 4 / kT; ++j) {
    const int u = tid + j * kT;
    __builtin_amdgcn_global_load_async_to_lds_b128(
        (GLB_AS wt_v4i*)(s4 + u), (LDS_AS wt_v4i*)(d4 + u), 0, 0);
  }
#else
  float4* d4 = (float4*)lds_dst;
  const float4* s4 = (const float4*)gsrc;
#pragma unroll
  for (int j = 0; j < kL / 4 / kT; ++j) {
    const int u = tid + j * kT;
    d4[u] = s4[u];
  }
#endif
}

__device__ __forceinline__ void row_wait() {
#if WT_TDM
  __builtin_amdgcn_s_wait_tensorcnt(0);
#elif WT_ASYNC
#if __has_builtin(__builtin_amdgcn_s_wait_asynccnt)
  __builtin_amdgcn_s_wait_asynccnt(0);
#else
  asm volatile("s_wait_asynccnt 0" ::: "memory");
#endif
#endif
}

// Haar analysis step: src (len 2n) -> dlo, dhi (len n).  float2-unit strided.
// y_lo[t] = wl0*src[2t] + wl1*src[2t+1];  y_hi[t] = wh0*src[2t] + wh1*src[2t+1]
__device__ __forceinline__ void haar_down(const float* __restrict__ src,
                                          float* __restrict__ dlo,
                                          float* __restrict__ dhi,
                                          int n, float wl0, float wl1,
                                          float wh0, float wh1, int tid) {
  const int units = n >> 1;
  for (int u = tid; u < units; u += kT) {
    const float4 sv = ((const float4*)src)[u];
    float2 lo, hi;
    lo.x = wl0 * sv.x + wl1 * sv.y;  lo.y = wl0 * sv.z + wl1 * sv.w;
    hi.x = wh0 * sv.x + wh1 * sv.y;  hi.y = wh0 * sv.z + wh1 * sv.w;
    ((float2*)dlo)[u] = lo;
    ((float2*)dhi)[u] = hi;
  }
}

// 4 outputs of a K=5 'same' conv (cross-correlation), scaled.
// out[t] = s * sum_k w[k]*in[t+k-2], zero-padded.
__device__ __forceinline__ float4 conv5_f4(const float* __restrict__ buf,
                                           int units, int u,
                                           float w0, float w1, float w2,
                                           float w3, float w4, float s) {
  const float4 c = ((const float4*)buf)[u];
  float2 lf = make_float2(0.f, 0.f), rt = make_float2(0.f, 0.f);
  if (u > 0)         lf = ((const float2*)buf)[2 * u - 1];
  if (u < units - 1) rt = ((const float2*)buf)[2 * u + 2];
  float4 o;
  o.x = w0 * lf.x + w1 * lf.y + w2 * c.x + w3 * c.y + w4 * c.z;
  o.y = w0 * lf.y + w1 * c.x  + w2 * c.y + w3 * c.z + w4 * c.w;
  o.z = w0 * c.x  + w1 * c.y  + w2 * c.z + w3 * c.w + w4 * rt.x;
  o.w = w0 * c.y  + w1 * c.z  + w2 * c.w + w3 * rt.x + w4 * rt.y;
  o.x *= s; o.y *= s; o.z *= s; o.w *= s;
  return o;
}

// In-place K=5 conv on two sibling buffers (low/high band of one level).
// Read everything into registers, barrier, write back (in-place safe).
template <int NU>
__device__ __forceinline__ void conv_pair(float* a, float* b, int n,
                                          const float* wa, float sa,
                                          const float* wb, float sb, int tid) {
  const int units = n >> 2;
  float4 ra[NU], rb[NU];
#pragma unroll
  for (int j = 0; j < NU; ++j) {
    const int u = tid + j * kT;
    ra[j] = conv5_f4(a, units, u, wa[0], wa[1], wa[2], wa[3], wa[4], sa);
    rb[j] = conv5_f4(b, units, u, wb[0], wb[1], wb[2], wb[3], wb[4], sb);
  }
  __syncthreads();
#pragma unroll
  for (int j = 0; j < NU; ++j) {
    const int u = tid + j * kT;
    ((float4*)a)[u] = ra[j];
    ((float4*)b)[u] = rb[j];
  }
  __syncthreads();
}

// Inverse Haar accumulate: tgt[2t]   += rl0*ll[t] + rh0*hh[t]
//                          tgt[2t+1] += rl1*ll[t] + rh1*hh[t]   (t in [0,n))
__device__ __forceinline__ void idwt_acc(const float* __restrict__ ll,
                                         const float* __restrict__ hh,
                                         float* __restrict__ tgt, int n,
                                         float rl0, float rl1,
                                         float rh0, float rh1, int tid) {
  const int units = n >> 1;
  for (int u = tid; u < units; u += kT) {
    const float2 lv = ((const float2*)ll)[u];
    const float2 hv = ((const float2*)hh)[u];
    float4 tv = ((float4*)tgt)[u];
    tv.x += rl0 * lv.x + rh0 * hv.x;
    tv.y += rl1 * lv.x + rh1 * hv.x;
    tv.z += rl0 * lv.y + rh0 * hv.y;
    tv.w += rl1 * lv.y + rh1 * hv.y;
    ((float4*)tgt)[u] = tv;
  }
}

// ---------------------------------------------------------------------------
__global__ void __launch_bounds__(kT) wtconv1d_fused(
    const float* __restrict__ x,      // (B,C,L)
    const float* __restrict__ wtf,    // (2C,1,2)
    const float* __restrict__ iwtf,   // (2C,1,2)
    const float* __restrict__ basew,  // (C,1,5)
    const float* __restrict__ baseb,  // (C)
    const float* __restrict__ basesc, // (1,C,1)
    const float* __restrict__ wavw,   // (3,2C,1,5)
    const float* __restrict__ wavs,   // (3,1,2C,1)
    float* __restrict__ out,          // (B,C,L)
    int nrows) {
  __shared__ float sm[kLdsFloats];
  const int tid = (int)threadIdx.x;
  const int G   = (int)gridDim.x;
  int r = (int)blockIdx.x;
  if (r >= nrows) return;

  // Kick off the first row's global->LDS DMA.
  row_load(sm + oX0, x + (size_t)r * kL, tid);

  int pb = 0;
  for (; r < nrows; r += G, pb ^= 1) {
    float* X   = sm + (pb ? oX1 : oX0);
    float* Xn  = sm + (pb ? oX0 : oX1);
    float* A   = sm + oA;
    float* BH  = sm + oB;
    float* C2L = sm + oC2L;
    float* C2H = sm + oC2H;
    float* D3L = sm + oD3L;
    float* D3H = sm + oD3H;

    const int c = r % kC;  // channel of this row

    // --- uniform per-channel weights (scalar loads) ---
    const float wl0 = wtf[(2 * c + 0) * 2 + 0], wl1 = wtf[(2 * c + 0) * 2 + 1];
    const float wh0 = wtf[(2 * c + 1) * 2 + 0], wh1 = wtf[(2 * c + 1) * 2 + 1];
    const float rl0 = iwtf[(2 * c + 0) * 2 + 0], rl1 = iwtf[(2 * c + 0) * 2 + 1];
    const float rh0 = iwtf[(2 * c + 1) * 2 + 0], rh1 = iwtf[(2 * c + 1) * 2 + 1];
    float cl[3][5], ch[3][5], sl[3], sh[3];
#pragma unroll
    for (int l = 0; l < 3; ++l) {
#pragma unroll
      for (int k = 0; k < 5; ++k) {
        cl[l][k] = wavw[(l * 2 * kC + 2 * c + 0) * 5 + k];
        ch[l][k] = wavw[(l * 2 * kC + 2 * c + 1) * 5 + k];
      }
      sl[l] = wavs[l * 2 * kC + 2 * c + 0];
      sh[l] = wavs[l * 2 * kC + 2 * c + 1];
    }
    float bw0 = basew[c * 5 + 0], bw1 = basew[c * 5 + 1], bw2 = basew[c * 5 + 2];
    float bw3 = basew[c * 5 + 3], bw4 = basew[c * 5 + 4];
    const float bbv = baseb[c];
    const float bsv = basesc[c];

    // Current row resident in LDS?
    row_wait();
    __syncthreads();

    // Prefetch the next row into the other X buffer; the TDM engine runs it
    // in the background while we compute (1 WG/WGP occupancy by design).
    const int rn = r + G;
    if (rn < nrows) row_load(Xn, x + (size_t)rn * kL, tid);

    // --- analysis: 3-level Haar cascade ---
    haar_down(X,   A,   BH,  kL1, wl0, wl1, wh0, wh1, tid); __syncthreads();
    haar_down(A,   C2L, C2H, kL2, wl0, wl1, wh0, wh1, tid); __syncthreads();
    haar_down(C2L, D3L, D3H, kL3, wl0, wl1, wh0, wh1, tid); __syncthreads();

    // --- per-level depthwise K=5 convs (+scale), in place ---
    conv_pair<kL3 / 4 / kT>(D3L, D3H, kL3, cl[2], sl[2], ch[2], sh[2], tid);
    conv_pair<kL2 / 4 / kT>(C2L, C2H, kL2, cl[1], sl[1], ch[1], sh[1], tid);
    conv_pair<kL1 / 4 / kT>(A,   BH,  kL1, cl[0], sl[0], ch[0], sh[0], tid);

    // --- synthesis: accumulate inverse Haar into the next-lower low band ---
    idwt_acc(D3L, D3H, C2L, kL3, rl0, rl1, rh0, rh1, tid); __syncthreads();
    idwt_acc(C2L, C2H, A,   kL2, rl0, rl1, rh0, rh1, tid); __syncthreads();

    // --- final: base conv on raw X + last inverse level, store to HBM ---
    {
      const float4* X4 = (const float4*)X;
      const float2* X2 = (const float2*)X;
      const float2* A2 = (const float2*)A;
      const float2* B2 = (const float2*)BH;
      wt_f32x4* o4 = (wt_f32x4*)(out + (size_t)r * kL);
      const int units = kL / 4;
      for (int u = tid; u < units; u += kT) {
        const float4 xc = X4[u];
        float2 xl = make_float2(0.f, 0.f), xr = make_float2(0.f, 0.f);
        if (u > 0)         xl = X2[2 * u - 1];
        if (u < units - 1) xr = X2[2 * u + 2];
        const float b0 = bw0 * xl.x + bw1 * xl.y + bw2 * xc.x + bw3 * xc.y + bw4 * xc.z;
        const float b1 = bw0 * xl.y + bw1 * xc.x + bw2 * xc.y + bw3 * xc.z + bw4 * xc.w;
        const float b2 = bw0 * xc.x + bw1 * xc.y + bw2 * xc.z + bw3 * xc.w + bw4 * xr.x;
        const float b3 = bw0 * xc.y + bw1 * xc.z + bw2 * xc.w + bw3 * xr.x + bw4 * xr.y;
        const float2 lv = A2[u];
        const float2 hv = B2[u];
        wt_f32x4 o = {
            (b0 + bbv) * bsv + rl0 * lv.x + rh0 * hv.x,
            (b1 + bbv) * bsv + rl1 * lv.x + rh1 * hv.x,
            (b2 + bbv) * bsv + rl0 * lv.y + rh0 * hv.y,
            (b3 + bbv) * bsv + rl1 * lv.y + rh1 * hv.y};
        // write-once output: non-temporal store keeps L2 clean
        __builtin_nontemporal_store(o, o4 + u);
      }
    }
    // loop-top wait+barrier separates this iteration's LDS reads from the
    // next iteration's LDS writes.
  }
}

// ---------------------------------------------------------------------------
extern "C" void kernel_launch(void* const* d_in, const int* in_sizes, int n_in,
                              void* d_out, int out_size, void* d_ws, size_t ws_size,
                              hipStream_t stream) {
  const float* x    = (const float*)d_in[0];
  const float* wtf  = (const float*)d_in[1];
  const float* iwtf = (const float*)d_in[2];
  const float* bw   = (const float*)d_in[3];
  const float* bb   = (const float*)d_in[4];
  const float* bsc  = (const float*)d_in[5];
  const float* ww   = (const float*)d_in[6];
  const float* wvs  = (const float*)d_in[7];
  float* out = (float*)d_out;

  const int nrows = in_sizes[0] / kL;          // B*C = 4096
  int grid = nrows < 512 ? nrows : 512;        // grid-stride: 8 rows/WG -> 7/8
  if (grid < 1) grid = 1;                      // of the HBM loads hidden

  wtconv1d_fused<<<dim3(grid), dim3(kT), 0, stream>>>(
      x, wtf, iwtf, bw, bb, bsc, ww, wvs, out, nrows);
}